// TripletLossWithHardMining_89215060673278
// MI455X (gfx1250) — compile-verified
//
#include <hip/hip_runtime.h>
#include <math.h>

// ---------------------------------------------------------------------------
// TripletLossWithHardMining for MI455X (gfx1250, wave32, WMMA).
//
// Distance matrices computed as GEMMs on the fp32 WMMA pipe:
//   ||a_i - y_j + eps||^2 = ||a_i||^2 + ||y_j||^2 + D*eps^2
//                           - 2<a_i,y_j> + 2*eps*(sum(a_i) - sum(y_j))
// fp32 WMMA (16x16x4) keeps the masked argmax/argmin selection exact.
// K-outer loop with 4 accumulators shares one A fragment across 4 column
// tiles per wave: 5 loads + 4 WMMAs per K-step.
// ---------------------------------------------------------------------------

#define BB 512
#define DD 256
static constexpr float EPS_F    = 1e-6f;
static constexpr float MARGIN_F = 0.5f;

typedef __attribute__((ext_vector_type(2))) float v2f;
typedef __attribute__((ext_vector_type(8))) float v8f;

// Workspace layout (floats)
#define WS_SUM_A   0
#define WS_N2_A    512
#define WS_SUM_P   1024
#define WS_N2_P    1536
#define WS_SUM_N   2048
#define WS_N2_N    2560
#define WS_DAP     3072
#define WS_DAN     3584
#define WS_VAL_N   4096
#define WS_VAL_P   4608
// total: 5120 floats = 20 KB

// ---------------------------------------------------------------------------
// Kernel A: per-row stats. One block per row, 256 threads = 256 columns.
// ---------------------------------------------------------------------------
__global__ __launch_bounds__(256)
void row_stats_kernel(const float* __restrict__ A,
                      const float* __restrict__ P,
                      const float* __restrict__ N,
                      float* __restrict__ ws) {
    const int i = blockIdx.x;
    const int t = threadIdx.x;

    const float a = A[(size_t)i * DD + t];
    const float p = P[(size_t)i * DD + t];
    const float n = N[(size_t)i * DD + t];

    const float dap = a - p + EPS_F;
    const float dan = a - n + EPS_F;

    float vals[8] = { a, a * a, p, p * p, n, n * n, dap * dap, dan * dan };
    float acc[8];

    __shared__ float red[256];
    #pragma unroll
    for (int q = 0; q < 8; ++q) {
        red[t] = vals[q];
        __syncthreads();
        for (int s = 128; s >= 1; s >>= 1) {
            if (t < s) red[t] += red[t + s];
            __syncthreads();
        }
        acc[q] = red[0];
        __syncthreads();
    }

    if (t == 0) {
        ws[WS_SUM_A + i] = acc[0];
        ws[WS_N2_A  + i] = acc[1];
        ws[WS_SUM_P + i] = acc[2];
        ws[WS_N2_P  + i] = acc[3];
        ws[WS_SUM_N + i] = acc[4];
        ws[WS_N2_N  + i] = acc[5];
        ws[WS_DAP   + i] = sqrtf(acc[6]);
        ws[WS_DAN   + i] = sqrtf(acc[7]);
    }
}

// ---------------------------------------------------------------------------
// Kernel B: WMMA distance row-block (16 rows x 512 cols) + masked hard mining.
// grid = (32 row-blocks, 2 cases). case 0: anchor vs negative, mask d<thr,
// argmax. case 1: anchor vs positive, mask d>thr, argmin.
// 256 threads = 8 wave32s. Each wave owns 4 column tiles of 16; one shared
// A fragment feeds 4 V_WMMA_F32_16X16X4_F32 accumulators per K-step.
// ---------------------------------------------------------------------------
__global__ __launch_bounds__(256)
void hard_mine_kernel(const float* __restrict__ A,
                      const float* __restrict__ P,
                      const float* __restrict__ N,
                      float* __restrict__ ws) {
    __shared__ float dist[16 * BB];   // 32 KB of the 320 KB WGP LDS

    const int  caseId = blockIdx.y;               // 0 = negative, 1 = positive
    const float* Y     = caseId ? P : N;
    const float* sum_y = ws + (caseId ? WS_SUM_P : WS_SUM_N);
    const float* n2_y  = ws + (caseId ? WS_N2_P  : WS_N2_N);
    const float* thr   = ws + (caseId ? WS_DAN   : WS_DAP);
    float*       out_v = ws + (caseId ? WS_VAL_P : WS_VAL_N);
    const float* sum_a = ws + WS_SUM_A;
    const float* n2_a  = ws + WS_N2_A;

    const int r0   = blockIdx.x * 16;
    const int tid  = threadIdx.x;
    const int wave = tid >> 5;                    // 0..7
    const int lane = tid & 31;

    // ISA VGPR layouts for V_WMMA_F32_16X16X4_F32:
    //  A (16x4):  lane m = lane&15, holds K = 2*(lane>>4) + {0,1}
    //  B (4x16):  lane n = lane&15, holds K = 2*(lane>>4) + {0,1} (B = Y^T)
    //  C (16x16): VGPR r -> M = r + 8*(lane>>4), N = lane&15
    const int mrow = lane & 15;
    const int kb   = (lane >> 4) << 1;
    const int half = lane >> 4;
    const float deps2 = (float)DD * EPS_F * EPS_F;

    const int c0base = wave * 64;                 // this wave's 64 columns
    const float* arow = A + (size_t)(r0 + mrow) * DD;
    const float* yrow0 = Y + (size_t)(c0base +  0 + mrow) * DD;
    const float* yrow1 = Y + (size_t)(c0base + 16 + mrow) * DD;
    const float* yrow2 = Y + (size_t)(c0base + 32 + mrow) * DD;
    const float* yrow3 = Y + (size_t)(c0base + 48 + mrow) * DD;

    v8f c0 = {}, c1 = {}, c2 = {}, c3 = {};
    #pragma unroll 4
    for (int kk = 0; kk < DD; kk += 4) {
        v2f av, b0, b1, b2, b3;
        av[0] = arow[kk + kb];   av[1] = arow[kk + kb + 1];
        b0[0] = yrow0[kk + kb];  b0[1] = yrow0[kk + kb + 1];
        b1[0] = yrow1[kk + kb];  b1[1] = yrow1[kk + kb + 1];
        b2[0] = yrow2[kk + kb];  b2[1] = yrow2[kk + kb + 1];
        b3[0] = yrow3[kk + kb];  b3[1] = yrow3[kk + kb + 1];
        c0 = __builtin_amdgcn_wmma_f32_16x16x4_f32(
                 false, av, false, b0, (short)0, c0, false, false);
        c1 = __builtin_amdgcn_wmma_f32_16x16x4_f32(
                 false, av, false, b1, (short)0, c1, false, false);
        c2 = __builtin_amdgcn_wmma_f32_16x16x4_f32(
                 false, av, false, b2, (short)0, c2, false, false);
        c3 = __builtin_amdgcn_wmma_f32_16x16x4_f32(
                 false, av, false, b3, (short)0, c3, false, false);
    }

    // Gram -> distance, park row-block in LDS
    {
        v8f acc[4] = { c0, c1, c2, c3 };
        const float aterm_base = deps2;
        #pragma unroll
        for (int t = 0; t < 4; ++t) {
            const int j = c0base + 16 * t + mrow;   // column (N = lane&15)
            const float yterm = n2_y[j] - 2.0f * EPS_F * sum_y[j];
            #pragma unroll
            for (int r = 0; r < 8; ++r) {
                const int row = r + 8 * half;
                const int i   = r0 + row;
                float sq = n2_a[i] + yterm + aterm_base - 2.0f * acc[t][r]
                           + 2.0f * EPS_F * sum_a[i];
                dist[row * BB + j] = sqrtf(fmaxf(sq, 0.0f));
            }
        }
    }
    __syncthreads();

    // Masked hard mining: wave w handles rows w and w+8.
    for (int rr = 0; rr < 2; ++rr) {
        const int row = wave + 8 * rr;
        const int i   = r0 + row;
        const float ti = thr[i];

        // argmax (case0) / argmin (case1) with first-occurrence tie-break
        float bestKey = -__builtin_inff();
        int   bestPos = 0x7FFFFFFF;
        #pragma unroll
        for (int s = 0; s < 16; ++s) {
            const int j = lane + 32 * s;
            const float v = dist[row * BB + j];
            const bool m = caseId ? (v > ti) : (v < ti);
            const float key = caseId ? -v : v;
            if (m && (key > bestKey || (key == bestKey && j < bestPos))) {
                bestKey = key; bestPos = j;
            }
        }
        #pragma unroll
        for (int off = 16; off >= 1; off >>= 1) {
            const float oKey = __shfl_xor(bestKey, off, 32);
            const int   oPos = __shfl_xor(bestPos, off, 32);
            if (oKey > bestKey || (oKey == bestKey && oPos < bestPos)) {
                bestKey = oKey; bestPos = oPos;
            }
        }
        const bool hasAny = (bestKey > -__builtin_inff());

        // compressed index: #masked positions <= bestPos, minus 1
        int cnt = 0;
        #pragma unroll
        for (int s = 0; s < 16; ++s) {
            const int j = lane + 32 * s;
            if (j <= bestPos) {
                const float v = dist[row * BB + j];
                cnt += caseId ? (v > ti) : (v < ti);
            }
        }
        #pragma unroll
        for (int off = 16; off >= 1; off >>= 1)
            cnt += __shfl_xor(cnt, off, 32);

        int jstar = cnt - 1;
        jstar = jstar < 0 ? 0 : (jstar > BB - 1 ? BB - 1 : jstar);
        const float val = hasAny ? dist[row * BB + jstar] : 0.0f;
        if (lane == 0) out_v[i] = val;
    }
}

// ---------------------------------------------------------------------------
// Kernel C: scalar means + hinge. One block, 512 threads, deterministic trees.
// ---------------------------------------------------------------------------
__global__ __launch_bounds__(512)
void loss_kernel(const float* __restrict__ ws, float* __restrict__ out) {
    __shared__ float red[512];
    __shared__ float s_dan, s_dpp;
    const int t = threadIdx.x;

    red[t] = ws[WS_VAL_N + t];
    __syncthreads();
    for (int s = 256; s >= 1; s >>= 1) {
        if (t < s) red[t] += red[t + s];
        __syncthreads();
    }
    if (t == 0) s_dan = red[0] / (float)BB;
    __syncthreads();

    red[t] = ws[WS_VAL_P + t];
    __syncthreads();
    for (int s = 256; s >= 1; s >>= 1) {
        if (t < s) red[t] += red[t + s];
        __syncthreads();
    }
    if (t == 0) s_dpp = red[0] / (float)BB;
    __syncthreads();

    const float term = ws[WS_DAP + t] - MARGIN_F * s_dpp - MARGIN_F * s_dan
                       + MARGIN_F;
    red[t] = fmaxf(term, 0.0f);
    __syncthreads();
    for (int s = 256; s >= 1; s >>= 1) {
        if (t < s) red[t] += red[t + s];
        __syncthreads();
    }
    if (t == 0) out[0] = red[0] / (float)BB;
}

// ---------------------------------------------------------------------------
extern "C" void kernel_launch(void* const* d_in, const int* in_sizes, int n_in,
                              void* d_out, int out_size, void* d_ws, size_t ws_size,
                              hipStream_t stream) {
    const float* A = (const float*)d_in[0];   // anchor   [512,256] f32
    const float* P = (const float*)d_in[1];   // positive [512,256] f32
    const float* N = (const float*)d_in[2];   // negative [512,256] f32
    float* ws  = (float*)d_ws;
    float* out = (float*)d_out;

    row_stats_kernel<<<BB, DD, 0, stream>>>(A, P, N, ws);
    hard_mine_kernel<<<dim3(32, 2), 256, 0, stream>>>(A, P, N, ws);
    loss_kernel<<<1, BB, 0, stream>>>(ws, out);
}